// AdaptiveSlotPruning_25563645346561
// MI455X (gfx1250) — compile-verified
//
#include <hip/hip_runtime.h>
#include <math.h>

// AdaptiveSlotPruning for MI455X (gfx1250, wave32).
// Bandwidth-bound: ~300MB traffic => ~9-13us floor at 23.3TB/s (masks' second
// read should hit the 192MB L2). WMMA f32 16x16x4 runs the gate MLP second
// layer (h[1536,16] @ W2[16,1]) exactly in f32.

typedef __attribute__((ext_vector_type(2))) float v2f;
typedef __attribute__((ext_vector_type(8))) float v8f;

#define B_  128
#define N_  16384
#define K_  12
#define D_  256
#define CHUNKS 32
#define ROWS_PER_CHUNK (N_ / CHUNKS)   // 512 rows of 12 floats per block
#define EPS_ 1e-8f

// d_out float offsets (outputs concatenated in return order)
#define OUT_SLOTS 0L
#define OUT_MASKS ((long)B_ * K_ * D_)                 // 393216
#define OUT_GATES (OUT_MASKS + (long)B_ * N_ * K_)     // 25559040
#define OUT_UTIL  (OUT_GATES + (long)B_ * K_)          // 25560576

// d_ws float offsets
#define WS_PARTIALS 0
#define WS_GATES    (B_ * CHUNKS * K_)                 // 49152 floats (~200KB total used)

// ---------------------------------------------------------------------------
// Pass 1: partial reduction of masks over N. Each block reduces 512 rows.
// Each thread owns 2 full K-rows (6 float4s) so the k-phase of every float4
// is compile-time constant (rows are exactly 3 float4s).
// ---------------------------------------------------------------------------
__global__ void asp_reduce(const float* __restrict__ masks,
                           float* __restrict__ partials) {
    const int blk = blockIdx.x;              // 0 .. B*CHUNKS-1
    const int b   = blk / CHUNKS;
    const int ch  = blk % CHUNKS;
    const int t   = threadIdx.x;             // 0..255

    const float4* src = (const float4*)masks
                      + ((long)b * N_ + (long)ch * ROWS_PER_CHUNK) * 3;
    const float4* p = src + (long)t * 6;     // 2 rows per thread

    const float4 v0 = p[0], v1 = p[1], v2 = p[2];
    const float4 v3 = p[3], v4 = p[4], v5 = p[5];

    float a[12];
    a[0] = v0.x + v3.x;  a[1]  = v0.y + v3.y;  a[2]  = v0.z + v3.z;  a[3]  = v0.w + v3.w;
    a[4] = v1.x + v4.x;  a[5]  = v1.y + v4.y;  a[6]  = v1.z + v4.z;  a[7]  = v1.w + v4.w;
    a[8] = v2.x + v5.x;  a[9]  = v2.y + v5.y;  a[10] = v2.z + v5.z;  a[11] = v2.w + v5.w;

    // wave32 butterfly reduction
    #pragma unroll
    for (int m = 16; m >= 1; m >>= 1) {
        #pragma unroll
        for (int j = 0; j < 12; ++j)
            a[j] += __shfl_xor(a[j], m, 32);
    }

    __shared__ float wred[8 * 12];           // 8 waves per block
    const int w    = t >> 5;
    const int lane = t & 31;
    if (lane == 0) {
        #pragma unroll
        for (int j = 0; j < 12; ++j) wred[w * 12 + j] = a[j];
    }
    __syncthreads();

    if (t < 12) {
        float s = 0.f;
        #pragma unroll
        for (int ww = 0; ww < 8; ++ww) s += wred[ww * 12 + t];
        partials[((long)b * CHUNKS + ch) * K_ + t] = s;
    }
}

// ---------------------------------------------------------------------------
// Pass 2: finish reduction, run gate MLP. One wave handles 16 (b,k) rows.
// Second MLP layer h[16rows,16] @ W2[16] done with 4 chained
// V_WMMA_F32_16X16X4_F32 (W2 broadcast into all 16 B-columns).
// Lane-half operand selection uses explicit 2-way ternaries on constant
// indices (single v_cndmask each) instead of dynamic register-array indexing.
// ---------------------------------------------------------------------------
__global__ void asp_gates(const float* __restrict__ partials,
                          const float* __restrict__ W1,
                          const float* __restrict__ b1,
                          const float* __restrict__ W2,
                          const float* __restrict__ b2,
                          float* __restrict__ gates_ws,
                          float* __restrict__ out) {
    const int tid  = blockIdx.x * blockDim.x + threadIdx.x;
    const int tile = tid >> 5;               // 0..95 (96 waves total)
    const int lane = threadIdx.x & 31;
    const int m    = lane & 15;
    const int r    = tile * 16 + m;          // (b,k) row, < 1536
    const int b    = r / K_;
    const int k    = r - b * K_;

    // finish the N-reduction for this row (32 chunk partials)
    float s = 0.f;
    #pragma unroll 8
    for (int c = 0; c < CHUNKS; ++c)
        s += partials[((long)b * CHUNKS + c) * K_ + k];
    const float u = s * (1.0f / (float)N_);

    if (lane < 16) out[OUT_UTIL + r] = u;    // utilization output

    // layer 1: h = relu(u * W1 + b1), all constant-indexed => registers
    float h[16], w2[16];
    #pragma unroll
    for (int j = 0; j < 16; ++j) {
        h[j]  = fmaxf(0.f, fmaf(u, W1[j], b1[j]));
        w2[j] = W2[j];
    }

    // layer 2 via WMMA: D = A(16x4) x B(4x16) + C, chained over K=16.
    // A layout (32-bit 16x4): lanes 0-15 hold K=k0,k0+1; lanes 16-31 K=k0+2,k0+3.
    const bool hi = (lane >= 16);
    v8f acc = {0.f, 0.f, 0.f, 0.f, 0.f, 0.f, 0.f, 0.f};
    #pragma unroll
    for (int i = 0; i < 4; ++i) {
        v2f av, bv;
        av.x = hi ? h[4 * i + 2]  : h[4 * i + 0];
        av.y = hi ? h[4 * i + 3]  : h[4 * i + 1];
        bv.x = hi ? w2[4 * i + 2] : w2[4 * i + 0];   // same value in every B column
        bv.y = hi ? w2[4 * i + 3] : w2[4 * i + 1];
        acc = __builtin_amdgcn_wmma_f32_16x16x4_f32(
            /*neg_a=*/false, av, /*neg_b=*/false, bv,
            /*c_mod=*/(short)0, acc, /*reuse_a=*/false, /*reuse_b=*/false);
    }

    // C/D layout: VGPR j, lanes 0-15 -> D[j][lane]; lanes 16-31 -> D[8+j][lane-16].
    // All columns identical, so lane 0 holds rows 0-7, lane 16 holds rows 8-15.
    const float bias = b2[0];
    if (lane == 0 || lane == 16) {
        const int base = tile * 16 + ((lane == 16) ? 8 : 0);
        #pragma unroll
        for (int j = 0; j < 8; ++j) {
            const float g = 1.0f / (1.0f + __expf(-(acc[j] + bias)));
            gates_ws[base + j]        = g;
            out[OUT_GATES + base + j] = g;
        }
    }
}

// ---------------------------------------------------------------------------
// Pass 3a: pruned_slots = slots * gates  (float4; gate idx = f4idx >> 6, D=256)
// ---------------------------------------------------------------------------
__global__ void asp_slots(const float* __restrict__ slots,
                          const float* __restrict__ gates_ws,
                          float* __restrict__ out) {
    const long i4 = (long)blockIdx.x * blockDim.x + threadIdx.x;  // < 98304
    const float4 v = ((const float4*)slots)[i4];
    const float g  = gates_ws[i4 >> 6];
    float4 o;
    o.x = v.x * g; o.y = v.y * g; o.z = v.z * g; o.w = v.w * g;
    ((float4*)(out + OUT_SLOTS))[i4] = o;
}

// ---------------------------------------------------------------------------
// Pass 3b: pruned_masks = normalize(masks * gates) per (b,n) row of 12.
// One thread per row: 3 float4 loads/stores. Each 256-thread block sits
// entirely inside one b (16384 % 256 == 0), so gates staged once in LDS.
// ---------------------------------------------------------------------------
__global__ void asp_masks(const float* __restrict__ masks,
                          const float* __restrict__ gates_ws,
                          float* __restrict__ out) {
    __shared__ float g[12];
    const long row = (long)blockIdx.x * 256 + threadIdx.x;  // < B*N
    const int  b   = (int)(row >> 14);                      // N = 16384
    if (threadIdx.x < 12) g[threadIdx.x] = gates_ws[b * K_ + threadIdx.x];
    __syncthreads();

    const float4* src = (const float4*)masks + row * 3;
    float4 m0 = src[0], m1 = src[1], m2 = src[2];

    m0.x *= g[0];  m0.y *= g[1];  m0.z *= g[2];  m0.w *= g[3];
    m1.x *= g[4];  m1.y *= g[5];  m1.z *= g[6];  m1.w *= g[7];
    m2.x *= g[8];  m2.y *= g[9];  m2.z *= g[10]; m2.w *= g[11];

    const float sum = (m0.x + m0.y + m0.z + m0.w)
                    + (m1.x + m1.y + m1.z + m1.w)
                    + (m2.x + m2.y + m2.z + m2.w);
    const float inv = 1.0f / (sum + EPS_);

    m0.x *= inv; m0.y *= inv; m0.z *= inv; m0.w *= inv;
    m1.x *= inv; m1.y *= inv; m1.z *= inv; m1.w *= inv;
    m2.x *= inv; m2.y *= inv; m2.z *= inv; m2.w *= inv;

    float4* dst = (float4*)(out + OUT_MASKS) + row * 3;
    dst[0] = m0; dst[1] = m1; dst[2] = m2;
}

// ---------------------------------------------------------------------------
extern "C" void kernel_launch(void* const* d_in, const int* in_sizes, int n_in,
                              void* d_out, int out_size, void* d_ws, size_t ws_size,
                              hipStream_t stream) {
    const float* slots = (const float*)d_in[0];
    const float* masks = (const float*)d_in[1];
    const float* W1    = (const float*)d_in[2];
    const float* b1    = (const float*)d_in[3];
    const float* W2    = (const float*)d_in[4];
    const float* b2    = (const float*)d_in[5];
    float* out = (float*)d_out;
    float* ws  = (float*)d_ws;

    float* partials = ws + WS_PARTIALS;   // B*CHUNKS*K floats, fully overwritten
    float* gates    = ws + WS_GATES;      // B*K floats, fully overwritten

    asp_reduce<<<B_ * CHUNKS, 256, 0, stream>>>(masks, partials);
    asp_gates<<<12, 256, 0, stream>>>(partials, W1, b1, W2, b2, gates, out);  // 96 waves
    asp_slots<<<(B_ * K_ * D_ / 4) / 256, 256, 0, stream>>>(slots, gates, out);
    asp_masks<<<(B_ * N_) / 256, 256, 0, stream>>>(masks, gates, out);
}